// Block_74225624809892
// MI455X (gfx1250) — compile-verified
//
#include <hip/hip_runtime.h>
#include <math.h>

// ---------------- static problem config (matches reference) ----------------
#define B_    2
#define N_    300
#define C_    256
#define H_    8
#define Dh    32
#define L_    13125
#define W1_   10000
#define W2_   12500
#define SCALE_ 0.17677669529663687f   // 32^-0.5
#define EPS_   1e-5f

typedef __attribute__((ext_vector_type(2))) float v2f;
typedef __attribute__((ext_vector_type(8))) float v8f;

__device__ __forceinline__ float gelu_exact(float x) {
    return 0.5f * x * (1.0f + erff(x * 0.70710678118654752f));
}

// ===========================================================================
// WMMA fp32 GEMM:  Y[M,N] = act(X[M,K] @ W[K,N] + bias)
// One wave32 computes one 16x16 tile with V_WMMA_F32_16X16X4_F32,
// accumulating over K in chunks of 4.  8 waves / 256-thread block.
// K and N are compile-time so all loads use immediate offsets off two base
// pointers; A rows are CLAMPED (not predicated) so the inner loop has
// uniform control flow and unconditional b64/b32 loads.
// A layout (16x4): lanes 0-15 hold M=lane, VGPR0/1 = K0/K1; lanes 16-31 K2/K3.
// B layout (4x16): lanes 0-15 hold N=lane, VGPR0/1 = K0/K1; lanes 16-31 K2/K3.
// C/D layout: VGPR r -> row M=r (lanes 0-15) / M=8+r (lanes 16-31), N=lane&15.
// ===========================================================================
template<int ACT, int K, int N>   // ACT: 0 = none, 2 = exact gelu
__global__ __launch_bounds__(256) void gemm_wmma(
        const float* __restrict__ X, const float* __restrict__ W,
        const float* __restrict__ bias, float* __restrict__ Y, int M)
{
    const int lane = threadIdx.x & 31;
    const int wid  = threadIdx.x >> 5;
    constexpr int ntiles = N >> 4;
    const int mtiles = (M + 15) >> 4;
    const int tile = blockIdx.x * 8 + wid;
    if (tile >= ntiles * mtiles) return;          // wave-uniform branch
    const int mt = tile / ntiles, nt = tile % ntiles;
    const int m0 = mt << 4, n0 = nt << 4;

    const int ka   = (lane >> 4) << 1;            // 0 or 2
    const int arow = min(m0 + (lane & 15), M - 1);  // clamp: uniform flow
    const int bcol = n0 + (lane & 15);
    const float* __restrict__ Xr = X + (size_t)arow * K + ka;  // 8B aligned
    const float* __restrict__ Wc = W + (size_t)ka * N + bcol;

    v8f acc = {0.f, 0.f, 0.f, 0.f, 0.f, 0.f, 0.f, 0.f};
    #pragma unroll 16
    for (int k0 = 0; k0 < K; k0 += 4) {
        v2f a = *(const v2f*)(Xr + k0);           // global_load_b64
        v2f b;
        b.x = Wc[(size_t)k0 * N];
        b.y = Wc[(size_t)k0 * N + N];
        acc = __builtin_amdgcn_wmma_f32_16x16x4_f32(
                  false, a, false, b, (short)0, acc, false, false);
    }

    const float bv = bias ? bias[bcol] : 0.f;
    const int mb = m0 + ((lane >> 4) << 3);
    float* __restrict__ Yc = Y + (size_t)mb * N + bcol;
    if (m0 + 16 <= M) {
        #pragma unroll
        for (int r = 0; r < 8; ++r) {
            float v = acc[r] + bv;
            if (ACT == 2) v = gelu_exact(v);
            Yc[(size_t)r * N] = v;
        }
    } else {
        #pragma unroll
        for (int r = 0; r < 8; ++r) {
            if (mb + r < M) {
                float v = acc[r] + bv;
                if (ACT == 2) v = gelu_exact(v);
                Yc[(size_t)r * N] = v;
            }
        }
    }
}

// ===========================================================================
// Fused attention: softmax(q.k^T * scale) @ v  per (b,h,n) row.
// Online (flash) softmax, 256 threads stride over L; K/V rows stream from L2
// (k_proj+v_proj = 54MB < 192MB L2).  LDS combine of 256 partials.
// ===========================================================================
__global__ __launch_bounds__(256) void attn_flash(
        const float* __restrict__ qp, const float* __restrict__ kp,
        const float* __restrict__ vp, float* __restrict__ xout)
{
    const int n = blockIdx.x, h = blockIdx.y, b = blockIdx.z;
    const int t = threadIdx.x;

    __shared__ float qsh[Dh];
    __shared__ float Mv[256];
    __shared__ float A[256];
    __shared__ float ACC[256 * Dh];

    const float* qrow = qp + ((size_t)(b * N_ + n)) * C_ + h * Dh;
    if (t < Dh) qsh[t] = qrow[t];
    __syncthreads();

    float qr[Dh];
    #pragma unroll
    for (int d = 0; d < Dh; ++d) qr[d] = qsh[d];

    float m = -3.0e38f, ssum = 0.f;
    float acc[Dh];
    #pragma unroll
    for (int d = 0; d < Dh; ++d) acc[d] = 0.f;

    for (int l = t; l < L_; l += 256) {
        const size_t roff = ((size_t)(b * L_ + l)) * C_ + h * Dh;
        const float4* kr = (const float4*)(kp + roff);
        const float4* vr = (const float4*)(vp + roff);
        if (l + 256 < L_) {
            __builtin_prefetch(kp + roff + 256 * (size_t)C_, 0, 0);
            __builtin_prefetch(vp + roff + 256 * (size_t)C_, 0, 0);
        }
        float s = 0.f;
        #pragma unroll
        for (int i = 0; i < 8; ++i) {
            float4 kv = kr[i];
            s += qr[4*i+0]*kv.x + qr[4*i+1]*kv.y + qr[4*i+2]*kv.z + qr[4*i+3]*kv.w;
        }
        s *= SCALE_;
        float mn   = fmaxf(m, s);
        float corr = __expf(m - mn);
        float p    = __expf(s - mn);
        ssum = ssum * corr + p;
        #pragma unroll
        for (int i = 0; i < 8; ++i) {
            float4 vv = vr[i];
            acc[4*i+0] = acc[4*i+0]*corr + p*vv.x;
            acc[4*i+1] = acc[4*i+1]*corr + p*vv.y;
            acc[4*i+2] = acc[4*i+2]*corr + p*vv.z;
            acc[4*i+3] = acc[4*i+3]*corr + p*vv.w;
        }
        m = mn;
    }

    // ---- combine 256 partials ----
    #pragma unroll
    for (int d = 0; d < Dh; ++d) ACC[t * Dh + d] = acc[d];
    A[t] = m;
    __syncthreads();
    for (int off = 128; off; off >>= 1) {
        if (t < off) A[t] = fmaxf(A[t], A[t + off]);
        __syncthreads();
    }
    const float gm = A[0];
    __syncthreads();
    const float w = __expf(m - gm);
    Mv[t] = w;
    A[t]  = w * ssum;
    __syncthreads();
    for (int off = 128; off; off >>= 1) {
        if (t < off) A[t] += A[t + off];
        __syncthreads();
    }
    const float tot = A[0];

    if (t < Dh) {
        float o = 0.f;
        for (int tt = 0; tt < 256; ++tt) o += Mv[tt] * ACC[tt * Dh + t];
        xout[((size_t)(b * N_ + n)) * C_ + h * Dh + t] = o / tot;
    }
}

// ===========================================================================
// Mask head, coarse grids: recompute 8 head logits at l (cheap, K row in L2),
// apply 8x8 linear + relu.  f2s layout (B*N, 8, 2500), f3s (B*N, 8, 625).
// ===========================================================================
__device__ __forceinline__ void head_logits(const float* __restrict__ qrow,
                                            const float* __restrict__ krow,
                                            float s[H_])
{
    #pragma unroll
    for (int h = 0; h < H_; ++h) {
        const float4* q4 = (const float4*)(qrow + h * Dh);
        const float4* k4 = (const float4*)(krow + h * Dh);
        float a = 0.f;
        #pragma unroll
        for (int i = 0; i < 8; ++i) {
            float4 qv = q4[i], kv = k4[i];
            a += qv.x*kv.x + qv.y*kv.y + qv.z*kv.z + qv.w*kv.w;
        }
        s[h] = a * SCALE_;
    }
}

__global__ __launch_bounds__(256) void mask_small(
        const float* __restrict__ qp, const float* __restrict__ kp,
        const float* __restrict__ l2w, const float* __restrict__ l2b,
        const float* __restrict__ l3w, const float* __restrict__ l3b,
        float* __restrict__ f2s, float* __restrict__ f3s)
{
    const int idx = blockIdx.x * 256 + threadIdx.x;
    const int TOT = B_ * N_ * 3125;
    if (idx >= TOT) return;
    const int bn = idx / 3125, j = idx % 3125;
    const int b  = bn / N_;
    const bool r2 = (j < 2500);
    const int jj  = r2 ? j : j - 2500;
    const int l   = r2 ? (W1_ + jj) : (W2_ + jj);

    float s[H_];
    head_logits(qp + (size_t)bn * C_, kp + ((size_t)(b * L_ + l)) * C_, s);

    const float* Wt = r2 ? l2w : l3w;
    const float* bt = r2 ? l2b : l3b;
    const int stride = r2 ? 2500 : 625;
    float* dst = r2 ? (f2s + (size_t)bn * H_ * 2500 + jj)
                    : (f3s + (size_t)bn * H_ * 625  + jj);
    #pragma unroll
    for (int o = 0; o < H_; ++o) {
        float v = bt[o];
        #pragma unroll
        for (int h = 0; h < H_; ++h) v += s[h] * Wt[h * H_ + o];
        dst[(size_t)o * stride] = fmaxf(v, 0.f);
    }
}

// ===========================================================================
// Mask finalize: f1 logits at full res + half-pixel bilinear upsample of
// f2 (50x50->100x100) and f3 (25x25->100x100), then 24->1 linear + relu.
// ===========================================================================
__device__ __forceinline__ float bilin(const float* __restrict__ pl, int S,
                                       float fy, float fx)
{
    int y0 = (int)floorf(fy), x0 = (int)floorf(fx);
    float wy = fy - (float)y0, wx = fx - (float)x0;
    int y0c = min(max(y0, 0), S - 1),     x0c = min(max(x0, 0), S - 1);
    int y1c = min(max(y0 + 1, 0), S - 1), x1c = min(max(x0 + 1, 0), S - 1);
    float a = pl[y0c * S + x0c], b = pl[y0c * S + x1c];
    float c = pl[y1c * S + x0c], d = pl[y1c * S + x1c];
    return (1.f - wy) * ((1.f - wx) * a + wx * b)
         +        wy  * ((1.f - wx) * c + wx * d);
}

__global__ __launch_bounds__(256) void mask_final(
        const float* __restrict__ qp, const float* __restrict__ kp,
        const float* __restrict__ l1w, const float* __restrict__ l1b,
        const float* __restrict__ lw,  const float* __restrict__ lb,
        const float* __restrict__ f2s, const float* __restrict__ f3s,
        float* __restrict__ mout)
{
    const int idx = blockIdx.x * 256 + threadIdx.x;
    const int TOT = B_ * N_ * W1_;
    if (idx >= TOT) return;
    const int bn = idx / W1_, p = idx % W1_;
    const int b  = bn / N_;
    const int y  = p / 100, x = p % 100;

    float s[H_];
    head_logits(qp + (size_t)bn * C_, kp + ((size_t)(b * L_ + p)) * C_, s);

    float out = lb[0];
    #pragma unroll
    for (int o = 0; o < H_; ++o) {
        float v = l1b[o];
        #pragma unroll
        for (int h = 0; h < H_; ++h) v += s[h] * l1w[h * H_ + o];
        out += fmaxf(v, 0.f) * lw[o];
    }
    // f2: 50->100, scale 2, half-pixel: src = (d+0.5)/2 - 0.5
    {
        float fy = (y + 0.5f) * 0.5f - 0.5f, fx = (x + 0.5f) * 0.5f - 0.5f;
        const float* base = f2s + (size_t)bn * H_ * 2500;
        #pragma unroll
        for (int o = 0; o < H_; ++o)
            out += bilin(base + o * 2500, 50, fy, fx) * lw[H_ + o];
    }
    // f3: 25->100, scale 4
    {
        float fy = (y + 0.5f) * 0.25f - 0.5f, fx = (x + 0.5f) * 0.25f - 0.5f;
        const float* base = f3s + (size_t)bn * H_ * 625;
        #pragma unroll
        for (int o = 0; o < H_; ++o)
            out += bilin(base + o * 625, 25, fy, fx) * lw[2 * H_ + o];
    }
    mout[idx] = fmaxf(out, 0.f);
}

// ===========================================================================
// y = LayerNorm(a + r) * g + be   (one 256-thread block per 256-wide row)
// ===========================================================================
__global__ __launch_bounds__(256) void add_ln(
        const float* __restrict__ a, const float* __restrict__ r,
        const float* __restrict__ g, const float* __restrict__ be,
        float* __restrict__ y)
{
    const int row = blockIdx.x, c = threadIdx.x;
    __shared__ float red[256];
    const float x = a[(size_t)row * C_ + c] + r[(size_t)row * C_ + c];

    red[c] = x; __syncthreads();
    for (int off = 128; off; off >>= 1) {
        if (c < off) red[c] += red[c + off];
        __syncthreads();
    }
    const float mean = red[0] * (1.0f / C_);
    __syncthreads();
    const float dx = x - mean;
    red[c] = dx * dx; __syncthreads();
    for (int off = 128; off; off >>= 1) {
        if (c < off) red[c] += red[c + off];
        __syncthreads();
    }
    const float var = red[0] * (1.0f / C_);
    y[(size_t)row * C_ + c] = dx * rsqrtf(var + EPS_) * g[c] + be[c];
}

// ===========================================================================
// launch
// ===========================================================================
static inline int gemm_grid(int M, int N) {
    int tiles = ((M + 15) / 16) * (N / 16);
    return (tiles + 7) / 8;
}

extern "C" void kernel_launch(void* const* d_in, const int* in_sizes, int n_in,
                              void* d_out, int out_size, void* d_ws, size_t ws_size,
                              hipStream_t stream) {
    const float* query = (const float*)d_in[0];
    const float* key   = (const float*)d_in[1];
    const float* value = (const float*)d_in[2];
    const float* qw    = (const float*)d_in[3];
    const float* kw    = (const float*)d_in[4];
    const float* vw    = (const float*)d_in[5];
    const float* pw    = (const float*)d_in[6];
    const float* pb    = (const float*)d_in[7];
    const float* l1w   = (const float*)d_in[8];
    const float* l1b   = (const float*)d_in[9];
    const float* l2w   = (const float*)d_in[10];
    const float* l2b   = (const float*)d_in[11];
    const float* l3w   = (const float*)d_in[12];
    const float* l3b   = (const float*)d_in[13];
    const float* lw    = (const float*)d_in[14];
    const float* lb    = (const float*)d_in[15];
    const float* ln1g  = (const float*)d_in[16];
    const float* ln1b  = (const float*)d_in[17];
    const float* fc1w  = (const float*)d_in[18];
    const float* fc1b  = (const float*)d_in[19];
    const float* fc2w  = (const float*)d_in[20];
    const float* fc2b  = (const float*)d_in[21];
    const float* ln2g  = (const float*)d_in[22];
    const float* ln2b  = (const float*)d_in[23];

    // workspace layout (floats) -- total ~29.9M floats (~120 MB)
    float* ws  = (float*)d_ws;
    float* qp  = ws;                        // 600*256
    float* kp  = qp  + 153600;              // 26250*256
    float* vp  = kp  + 6720000;             // 26250*256
    float* ax  = vp  + 6720000;             // 600*256  attention out
    float* px  = ax  + 153600;              // 600*256  proj out
    float* q1  = px  + 153600;              // 600*256  after LN1
    float* hb  = q1  + 153600;              // 600*1024 gelu(fc1)
    float* m2  = hb  + 614400;              // 600*256  fc2 out
    float* f2s = m2  + 153600;              // 600*8*2500
    float* f3s = f2s + 12000000;            // 600*8*625

    const int BN = B_ * N_;      // 600
    const int BL = B_ * L_;      // 26250

    // ---- projections (WMMA fp32 GEMMs) ----
    gemm_wmma<0, 256, 256><<<gemm_grid(BN, C_), 256, 0, stream>>>(query, qw, nullptr, qp, BN);
    gemm_wmma<0, 256, 256><<<gemm_grid(BL, C_), 256, 0, stream>>>(key,   kw, nullptr, kp, BL);
    gemm_wmma<0, 256, 256><<<gemm_grid(BL, C_), 256, 0, stream>>>(value, vw, nullptr, vp, BL);

    // ---- fused flash attention (K/V stream from L2) ----
    attn_flash<<<dim3(N_, H_, B_), 256, 0, stream>>>(qp, kp, vp, ax);

    // ---- output projection + residual/LN + MLP + LN ----
    gemm_wmma<0, 256, 256><<<gemm_grid(BN, C_), 256, 0, stream>>>(ax, pw, pb, px, BN);
    add_ln<<<BN, 256, 0, stream>>>(query, px, ln1g, ln1b, q1);
    gemm_wmma<2, 256, 1024><<<gemm_grid(BN, 4 * C_), 256, 0, stream>>>(q1, fc1w, fc1b, hb, BN);
    gemm_wmma<0, 1024, 256><<<gemm_grid(BN, C_), 256, 0, stream>>>(hb, fc2w, fc2b, m2, BN);
    float* outq = (float*)d_out;
    add_ln<<<BN, 256, 0, stream>>>(q1, m2, ln2g, ln2b, outq);

    // ---- mask head ----
    mask_small<<<(BN * 3125 + 255) / 256, 256, 0, stream>>>(qp, kp, l2w, l2b, l3w, l3b, f2s, f3s);
    mask_final<<<(BN * W1_ + 255) / 256, 256, 0, stream>>>(qp, kp, l1w, l1b, lw, lb,
                                                           f2s, f3s, outq + 153600);
}